// CTCLoss_55585466745200
// MI455X (gfx1250) — compile-verified
//
#include <hip/hip_runtime.h>
#include <hip/hip_bf16.h>
#include <stdint.h>

// Problem constants (match reference)
#define T_    1024
#define B_    128
#define V_    512
#define LMAX_ 128
#define S_    257          // 2*LMAX+1
#define CH_   8            // timesteps per TDM tile
#define NEGV  (-1.0e9f)

typedef unsigned int u32x4 __attribute__((ext_vector_type(4)));
typedef unsigned int u32x8 __attribute__((ext_vector_type(8)));
typedef float        v8f   __attribute__((ext_vector_type(8)));
typedef float        v2f   __attribute__((ext_vector_type(2)));

// log(exp(a)+exp(b)), stable; with NEGV sentinel behaves like jnp.logaddexp
__device__ __forceinline__ float lae(float a, float b) {
    float m = fmaxf(a, b);
    float d = fabsf(a - b);
    return m + log1pf(expf(-d));
}

// ---------------------------------------------------------------------------
// Kernel A: per-row logsumexp of acts[T,B,V] -> lse[b*T + t]  (transposed for
// coalesced reads in the recursion kernel). One wave per row, 8 rows/block.
// ---------------------------------------------------------------------------
__global__ __launch_bounds__(256) void lse_kernel(const float* __restrict__ acts,
                                                  float* __restrict__ lse) {
    const int lane = threadIdx.x & 31;
    const int wv   = threadIdx.x >> 5;
    const int row  = blockIdx.x * 8 + wv;            // row = t*B + b
    const float4* rp = reinterpret_cast<const float4*>(acts + (size_t)row * V_);

    float vals[16];
    float mx = -3.4e38f;
#pragma unroll
    for (int j = 0; j < 4; ++j) {
        float4 v = rp[lane + 32 * j];
        vals[j * 4 + 0] = v.x; vals[j * 4 + 1] = v.y;
        vals[j * 4 + 2] = v.z; vals[j * 4 + 3] = v.w;
        mx = fmaxf(mx, fmaxf(fmaxf(v.x, v.y), fmaxf(v.z, v.w)));
    }
#pragma unroll
    for (int off = 16; off >= 1; off >>= 1) mx = fmaxf(mx, __shfl_xor(mx, off, 32));
    float s = 0.0f;
#pragma unroll
    for (int i = 0; i < 16; ++i) s += expf(vals[i] - mx);
#pragma unroll
    for (int off = 16; off >= 1; off >>= 1) s += __shfl_xor(s, off, 32);
    if (lane == 0) {
        int t = row >> 7;        // /B_
        int b = row & 127;       // %B_
        lse[b * T_ + t] = mx + logf(s);
    }
}

// ---------------------------------------------------------------------------
// TDM: async 2-D tile load  acts[t0..t0+CH-1][b][0..V-1]  -> LDS (contiguous).
// D# built per CDNA5 ISA ch.8; 2-operand form = VADDR2/3 NULL (2-D tensor).
// ---------------------------------------------------------------------------
__device__ __forceinline__ void tdm_load_rows(const float* acts, int b, int t0,
                                              uint32_t lds_off) {
    uint64_t ga = (uint64_t)(uintptr_t)acts +
                  (((uint64_t)t0 * B_ + (uint64_t)b) * V_) * 4ull;
    uint32_t dim1 = (uint32_t)(T_ - t0);   // real rows remaining (no OOB reads)

    u32x4 g0;
    g0[0] = 1u;                                   // count=1, user descriptor
    g0[1] = lds_off;                              // lds_addr (bytes)
    g0[2] = (uint32_t)ga;                         // global_addr[31:0]
    g0[3] = (uint32_t)((ga >> 32) & 0x1FFFFFFu)   // global_addr[56:32]
          | (2u << 30);                           // type = 2 ("image")

    u32x8 g1;
    g1[0] = (2u << 16);                           // data_size = 4 bytes
    g1[1] = (uint32_t)(V_ & 0xFFFF) << 16;        // tensor_dim0[15:0] @ bits 63:48
    g1[2] = (dim1 & 0xFFFFu) << 16;               // tensor_dim0 hi=0 | tensor_dim1 lo
    g1[3] = (dim1 >> 16) | ((uint32_t)V_ << 16);  // tensor_dim1 hi | tile_dim0=512
    g1[4] = (uint32_t)CH_;                        // tile_dim1=CH, tile_dim2=0
    g1[5] = (uint32_t)(B_ * V_);                  // tensor_dim0_stride = 65536 elems
    g1[6] = 0u;                                   // stride hi / dim1_stride (unused 2-D)
    g1[7] = 0u;

    asm volatile("tensor_load_to_lds %0, %1" :: "s"(g0), "s"(g1) : "memory");
}

// ---------------------------------------------------------------------------
// Kernel B: CTC alpha recursion. One wave per batch sample. Lane L owns states
// s = 8L..8L+7; lane 31 additionally owns s=256. Neighbor states come from
// __shfl_up (no barriers in the serial loop). Row tiles arrive via TDM
// double-buffering; LSE cached in LDS.
// ---------------------------------------------------------------------------
__global__ __launch_bounds__(32) void ctc_alpha_kernel(const float* __restrict__ acts,
                                                       const int* __restrict__ labels,
                                                       const int* __restrict__ act_lens,
                                                       const int* __restrict__ label_lens,
                                                       const float* __restrict__ lse,
                                                       float* __restrict__ loss_out) {
    __shared__ float buf[2][CH_ * V_];   // 32 KB: double-buffered row tiles
    __shared__ float lse_sh[T_];         // 4 KB
    __shared__ float afinal[S_ + 7];     // final alpha, padded

    const int b    = blockIdx.x;
    const int lane = threadIdx.x;

    // Build extended-label data for this lane's 9 states.
    int  ext_s[9];
    bool allow[9];
#pragma unroll
    for (int k = 0; k < 9; ++k) {
        int s = lane * 8 + k;            // lane31,k=8 -> 256; others' k=8 unused
        if (s & 1) {
            int li = s >> 1;
            int lv = labels[b * LMAX_ + li];
            ext_s[k] = lv;
            // allow2: ext[s]!=blank && ext[s]!=ext[s-2]; s==1 -> ext[-1]=blank
            allow[k] = (li == 0) ? true : (lv != labels[b * LMAX_ + li - 1]);
        } else {
            ext_s[k] = 0;                // blank
            allow[k] = false;
        }
    }
    const int alen = act_lens[b];
    const int llen = label_lens[b];

    // Cache all LSE values for this sample (coalesced: lse laid out [B][T]).
    for (int i = lane; i < T_; i += 32) lse_sh[i] = lse[b * T_ + i];
    __syncthreads();

    float alpha[9];
#pragma unroll
    for (int k = 0; k < 9; ++k) alpha[k] = NEGV;

    const int nch = (alen + CH_ - 1) / CH_;
    tdm_load_rows(acts, b, 0, (uint32_t)(uintptr_t)&buf[0][0]);

    for (int c = 0; c < nch; ++c) {
        if (c + 1 < nch) {
            tdm_load_rows(acts, b, (c + 1) * CH_,
                          (uint32_t)(uintptr_t)&buf[(c + 1) & 1][0]);
            __builtin_amdgcn_s_wait_tensorcnt(1);   // chunk c landed, c+1 in flight
        } else {
            __builtin_amdgcn_s_wait_tensorcnt(0);
        }
        const float* row = &buf[c & 1][0];
        const int t0   = c * CH_;
        const int tend = min(CH_, alen - t0);

        for (int tt = 0; tt < tend; ++tt) {
            const int t = t0 + tt;
            const float lsev = lse_sh[t];
            float lp[9];
#pragma unroll
            for (int k = 0; k < 9; ++k)
                lp[k] = row[tt * V_ + ext_s[k]] - lsev;

            if (t == 0) {
#pragma unroll
                for (int k = 0; k < 9; ++k) {
                    int s = lane * 8 + k;
                    alpha[k] = (s < 2) ? lp[k] : NEGV;
                }
            } else {
                float n1 = __shfl_up(alpha[7], 1, 32);   // alpha[s-1] from prev lane
                float n2 = __shfl_up(alpha[6], 1, 32);   // alpha[s-2] from prev lane
                if (lane == 0) { n1 = NEGV; n2 = NEGV; }
                float na[9];
#pragma unroll
                for (int k = 0; k < 9; ++k) {
                    float p0 = alpha[k];
                    float p1 = (k >= 1) ? alpha[k - 1] : n1;
                    float p2 = (k >= 2) ? alpha[k - 2] : ((k == 1) ? n1 : n2);
                    float a  = lae(p0, p1);
                    if (allow[k]) a = lae(a, p2);
                    na[k] = a + lp[k];
                }
#pragma unroll
                for (int k = 0; k < 9; ++k) alpha[k] = na[k];
            }
        }
    }

    // Publish final alpha and compute per-sample loss.
#pragma unroll
    for (int k = 0; k < 8; ++k) afinal[lane * 8 + k] = alpha[k];
    if (lane == 31) afinal[256] = alpha[8];
    __syncthreads();
    if (lane == 0) {
        int end  = 2 * llen;                 // last blank position (<= 256)
        float v1 = afinal[end];
        float v2 = (end >= 1) ? afinal[end - 1] : NEGV;
        loss_out[b] = -lae(v1, v2);
    }
}

// ---------------------------------------------------------------------------
// Kernel C: mean over the 128 per-sample losses using an exact f32 WMMA
// (A = pair-reduced losses in 16x4, B = ones -> D rows hold the row sums).
// ---------------------------------------------------------------------------
__global__ __launch_bounds__(32) void loss_mean_kernel(const float* __restrict__ loss,
                                                       float* __restrict__ out) {
    const int lane  = threadIdx.x;
    const int M     = lane & 15;
    const int kbase = (lane >> 4) * 2;       // lanes 0-15 -> K=0,1; 16-31 -> K=2,3

    // 128 -> 64 pairwise, filled into the 16x4 f32 A-operand layout.
    int i0 = (M * 4 + kbase) * 2;
    int i1 = (M * 4 + kbase + 1) * 2;
    v2f A;  A[0] = loss[i0] + loss[i0 + 1];
            A[1] = loss[i1] + loss[i1 + 1];
    v2f Bv; Bv[0] = 1.0f; Bv[1] = 1.0f;      // 4x16 all-ones
    v8f C = {};
    C = __builtin_amdgcn_wmma_f32_16x16x4_f32(false, A, false, Bv,
                                              (short)0, C, false, false);
    // Lanes 0-15 hold D[M=0..7][N=lane]; lanes 16-31 hold D[M=8..15][N].
    float s = C[0] + C[1] + C[2] + C[3] + C[4] + C[5] + C[6] + C[7];
    float tot = s + __shfl_xor(s, 16, 32);   // add the other M-half
    if (lane == 0) out[0] = tot * (1.0f / (float)B_);
}

// ---------------------------------------------------------------------------
extern "C" void kernel_launch(void* const* d_in, const int* in_sizes, int n_in,
                              void* d_out, int out_size, void* d_ws, size_t ws_size,
                              hipStream_t stream) {
    const float* acts       = (const float*)d_in[0];
    const int*   labels     = (const int*)d_in[1];
    const int*   act_lens   = (const int*)d_in[2];
    const int*   label_lens = (const int*)d_in[3];
    float*       out        = (float*)d_out;

    float* ws_lse  = (float*)d_ws;                 // [B][T] = 512 KB
    float* ws_loss = ws_lse + (size_t)B_ * T_;     // [B]

    lse_kernel<<<(T_ * B_) / 8, 256, 0, stream>>>(acts, ws_lse);
    ctc_alpha_kernel<<<B_, 32, 0, stream>>>(acts, labels, act_lens, label_lens,
                                            ws_lse, ws_loss);
    loss_mean_kernel<<<1, 32, 0, stream>>>(ws_loss, out);
}